// FFMP_39745627357786
// MI455X (gfx1250) — compile-verified
//
#include <hip/hip_runtime.h>
#include <math.h>

// ext vector types matching CDNA5 WMMA f32 16x16x4 fragments (wave32):
//   A 16x4 f32 : 64 vals / 32 lanes = 2 VGPRs -> float2 per lane
//   B 4x16 f32 : 2 VGPRs -> float2 per lane
//   C/D 16x16  : 8 VGPRs -> float8 per lane
typedef __attribute__((ext_vector_type(2))) float v2f;
typedef __attribute__((ext_vector_type(8))) float v8f;

#define N_FEAT 64
#define N_DIM  64
#define BATCH  128
#define NPAIRS 2080            // 64*65/2
#define TILES_PER_BATCH 130    // 2080 / 16
#define WAVES_PER_BLOCK 4
#define ROW_STRIDE 68          // 64 floats + 4 pad -> conflict-free LDS banks

__global__ __launch_bounds__(128)
void FFMP_39745627357786_kernel(const float* __restrict__ in,
                                const int*   __restrict__ x_idx,
                                const int*   __restrict__ y_idx,
                                float*       __restrict__ out)
{
    // Per-wave region: X[16][68] + Y[16][68] floats = 8704 B; 4 waves = 34 KB.
    __shared__ float lds[WAVES_PER_BLOCK * 2 * 16 * ROW_STRIDE];

    const int tid  = threadIdx.x;
    const int wave = tid >> 5;
    const int lane = tid & 31;
    const int hi   = lane >> 4;   // which 16-lane half
    const int lm   = lane & 15;

    const int tile = blockIdx.x * WAVES_PER_BLOCK + wave;  // 16640 tiles total
    const int b    = tile / TILES_PER_BATCH;
    const int k0   = (tile % TILES_PER_BATCH) * 16;        // first pair of tile

    float* X = &lds[wave * 2 * 16 * ROW_STRIDE];
    float* Y = X + 16 * ROW_STRIDE;

    // lane r (r = lane&15) caches the gather indices for pair k0+r
    const int xi_l = x_idx[k0 + lm];
    const int yi_l = y_idx[k0 + lm];

    const size_t in_base = (size_t)b * (N_FEAT * N_FEAT) * N_DIM;

    // ---- Stage 16 X rows + 16 Y rows into LDS, float4 granularity ----
    // iteration it: lanes 0-15 load row 2*it, lanes 16-31 row 2*it+1.
    // Input rows stay regular-temporal: the 134 MB input fits in the 192 MB L2,
    // and every row is touched twice (once as x, once as y).
#pragma unroll
    for (int it = 0; it < 8; ++it) {
        const int row = 2 * it + hi;
        const int xi  = __shfl(xi_l, row, 32);
        const int yi  = __shfl(yi_l, row, 32);
        const float4 xv = *(const float4*)(in + in_base + (size_t)xi * N_DIM + lm * 4);
        const float4 yv = *(const float4*)(in + in_base + (size_t)yi * N_DIM + lm * 4);
        *(float4*)&X[row * ROW_STRIDE + lm * 4] = xv;
        *(float4*)&Y[row * ROW_STRIDE + lm * 4] = yv;
    }

    __syncthreads();

    // ---- ip = diag(X.Y^T), dd = diag(D.D^T) via v_wmma_f32_16x16x4_f32 ----
    // A-frag (16x4) and B-frag (4x16, B[k][n] = Yrow[n][k]) are the SAME
    // per-lane float2: lane m (low) = [row m][4t+0..1], lane m (high) = [row m][4t+2..3].
    v8f acc_ip = {};
    v8f acc_dd = {};
#pragma unroll
    for (int t = 0; t < 16; ++t) {
        const int c = 4 * t + 2 * hi;
        const v2f ax = *(const v2f*)&X[lm * ROW_STRIDE + c];
        const v2f ay = *(const v2f*)&Y[lm * ROW_STRIDE + c];
        const v2f ad = ax - ay;
        // 8 args: (neg_a, A, neg_b, B, c_mod, C, reuse_a, reuse_b)
        acc_ip = __builtin_amdgcn_wmma_f32_16x16x4_f32(false, ax, false, ay,
                                                       (short)0, acc_ip, false, false);
        acc_dd = __builtin_amdgcn_wmma_f32_16x16x4_f32(false, ad, false, ad,
                                                       (short)0, acc_dd, false, false);
    }

    // Diagonal (p,p): VGPR p&7, lane p (p<8) or lane p+16 (p>=8)
    const bool havep = (lane < 8) || (lane >= 24);
    const int  p     = (lane < 8) ? lane : (lane - 16);   // pair index 0..15
    const int  r     = p & 7;
    float aip = acc_ip[0];
    float add = acc_dd[0];
#pragma unroll
    for (int q = 1; q < 8; ++q) {
        if (r == q) { aip = acc_ip[q]; add = acc_dd[q]; }
    }

    // ---- Elementwise s/d/p stream: float2, coalesced 256B segments ----
    // Output is a 206 MB write-only stream, never re-read by this kernel:
    // non-temporal stores keep it from evicting the L2-resident input.
    const size_t out_base = ((size_t)b * NPAIRS + k0) * 194;
#pragma unroll
    for (int row = 0; row < 16; ++row) {
        const v2f xv = *(const v2f*)&X[row * ROW_STRIDE + 2 * lane];
        const v2f yv = *(const v2f*)&Y[row * ROW_STRIDE + 2 * lane];
        const v2f s  = xv + yv;
        const v2f d  = xv - yv;
        const v2f pr = xv * yv;
        float* o = out + out_base + (size_t)row * 194 + 2 * lane;
        __builtin_nontemporal_store(s,  (v2f*)(o));        // [0..63]
        __builtin_nontemporal_store(d,  (v2f*)(o + 64));   // [64..127]
        __builtin_nontemporal_store(pr, (v2f*)(o + 128));  // [128..191]
    }

    if (havep) {
        float* o = out + out_base + (size_t)p * 194;
        __builtin_nontemporal_store(aip,        o + 192);
        __builtin_nontemporal_store(sqrtf(add), o + 193);
    }
}

extern "C" void kernel_launch(void* const* d_in, const int* in_sizes, int n_in,
                              void* d_out, int out_size, void* d_ws, size_t ws_size,
                              hipStream_t stream) {
    const float* in   = (const float*)d_in[0];
    const int*   xidx = (const int*)d_in[1];
    const int*   yidx = (const int*)d_in[2];
    float*       out  = (float*)d_out;

    const int total_tiles = BATCH * TILES_PER_BATCH;        // 16640 wave-tiles
    dim3 grid(total_tiles / WAVES_PER_BLOCK);               // 4160 blocks
    FFMP_39745627357786_kernel<<<grid, 128, 0, stream>>>(in, xidx, yidx, out);
}